// SimpleGCNModel_3470333575753
// MI455X (gfx1250) — compile-verified
//
#include <hip/hip_runtime.h>
#include <math.h>

typedef float v2f __attribute__((ext_vector_type(2)));
typedef float v8f __attribute__((ext_vector_type(8)));

#define IN_DIM 512
#define HID    64
#define SCAN_T 1024

// ---------------------------------------------------------------------------
// CSR build step 1: zero per-dst counters
// ---------------------------------------------------------------------------
__global__ void k_zero_cnt(int* __restrict__ cnt, int n) {
    int i = blockIdx.x * blockDim.x + threadIdx.x;
    if (i < n) cnt[i] = 0;
}

// CSR build step 2: count edges per dst (int atomics, 1.6M total)
__global__ void k_count(const int* __restrict__ ei, int* __restrict__ cnt, int E) {
    int e = blockIdx.x * blockDim.x + threadIdx.x;
    if (e < E) atomicAdd(&cnt[ei[E + e]], 1);
}

// CSR build step 3: exclusive prefix sum over cnt -> row_ptr[0..n]
// Single 1024-thread workgroup, Hillis-Steele per 1024-chunk with carry.
__global__ void k_scan(const int* __restrict__ cnt, int* __restrict__ row_ptr, int n) {
    __shared__ int sm[SCAN_T];
    __shared__ int carry_s;
    const int tid = threadIdx.x;
    if (tid == 0) carry_s = 0;
    __syncthreads();
    for (int base = 0; base < n; base += SCAN_T) {
        const int i = base + tid;
        const int v = (i < n) ? cnt[i] : 0;
        sm[tid] = v;
        __syncthreads();
        #pragma unroll
        for (int off = 1; off < SCAN_T; off <<= 1) {
            const int t = (tid >= off) ? sm[tid - off] : 0;
            __syncthreads();
            sm[tid] += t;
            __syncthreads();
        }
        const int incl  = sm[tid];
        const int carry = carry_s;
        if (i < n) row_ptr[i] = carry + incl - v;   // exclusive
        __syncthreads();
        if (tid == SCAN_T - 1) carry_s = carry + incl;
        __syncthreads();
    }
    if (tid == 0) row_ptr[n] = carry_s;             // == E
}

// CSR build step 4: dis[n] = rsqrt(cnt[n]+1)  (self-loop), and pos = row_ptr copy
__global__ void k_prep(const int* __restrict__ cnt, const int* __restrict__ row_ptr,
                       float* __restrict__ dis, int* __restrict__ pos, int n) {
    int i = blockIdx.x * blockDim.x + threadIdx.x;
    if (i < n) {
        dis[i] = rsqrtf((float)cnt[i] + 1.0f);
        pos[i] = row_ptr[i];
    }
}

// CSR build step 5: scatter src ids into dst-sorted order
__global__ void k_scatter(const int* __restrict__ ei, int* __restrict__ pos,
                          int* __restrict__ csr_src, int E) {
    int e = blockIdx.x * blockDim.x + threadIdx.x;
    if (e < E) {
        const int d = ei[E + e];
        const int p = atomicAdd(&pos[d], 1);
        csr_src[p] = ei[e];
    }
}

// ---------------------------------------------------------------------------
// GEMM1: xt = x @ W1  via V_WMMA_F32_16X16X4_F32 (exact f32).
// One wave -> 16 rows x all 64 cols (4 accumulators): x streamed exactly once.
// ---------------------------------------------------------------------------
__global__ void k_gemm1_wmma(const float* __restrict__ x,
                             const float* __restrict__ W1,
                             float* __restrict__ xt,
                             int nTiles) {
    const int wave = (int)((blockIdx.x * (long long)blockDim.x + threadIdx.x) >> 5);
    if (wave >= nTiles) return;                 // wave-uniform: EXEC stays all-1s
    const int lane = threadIdx.x & 31;
    const int m    = lane & 15;                 // A row / B,C col within tile
    const int kh   = lane >> 4;                 // K half-select (0 or 1)

    const long long row0 = (long long)wave * 16;
    const float* __restrict__ xrow = x + (row0 + m) * IN_DIM;

    v8f acc0 = {}, acc1 = {}, acc2 = {}, acc3 = {};

    for (int k = 0; k < IN_DIM; k += 4) {
        const int kb = k + 2 * kh;
        // A tile 16x4: vgpr v holds K = kb + v   (aligned float2 load)
        const float2 af = *(const float2*)(xrow + kb);
        v2f a; a.x = af.x; a.y = af.y;
        // B tile 4x16: lane col = m, vgpr v holds K = kb + v
        const float* __restrict__ w0 = W1 + (long long)kb * HID + m;  // row kb
        const float* __restrict__ w1 = w0 + HID;                      // row kb+1
        v2f b0; b0.x = w0[0];  b0.y = w1[0];
        v2f b1; b1.x = w0[16]; b1.y = w1[16];
        v2f b2; b2.x = w0[32]; b2.y = w1[32];
        v2f b3; b3.x = w0[48]; b3.y = w1[48];

        acc0 = __builtin_amdgcn_wmma_f32_16x16x4_f32(false, a, false, b0, (short)0, acc0, false, false);
        acc1 = __builtin_amdgcn_wmma_f32_16x16x4_f32(false, a, false, b1, (short)0, acc1, false, false);
        acc2 = __builtin_amdgcn_wmma_f32_16x16x4_f32(false, a, false, b2, (short)0, acc2, false, false);
        acc3 = __builtin_amdgcn_wmma_f32_16x16x4_f32(false, a, false, b3, (short)0, acc3, false, false);
    }

    // C/D layout: vgpr v -> row = row0 + v + 8*kh, col = m
    #pragma unroll
    for (int v = 0; v < 8; ++v) {
        const long long base = (row0 + v + 8 * kh) * HID + m;
        xt[base +  0] = acc0[v];
        xt[base + 16] = acc1[v];
        xt[base + 32] = acc2[v];
        xt[base + 48] = acc3[v];
    }
}

// ---------------------------------------------------------------------------
// Pull layer 1 (fused with layer-2 transform):
//   acc[n][:] = dis[n]^2 * xt[n][:] + sum_{s in N(n)} dis[s]*dis[n] * xt[s][:]
//   h2t[n]    = relu(acc + b1) @ W2          (64 -> 2, wave shfl reduction)
// One wave per node; lane holds features {2*lane, 2*lane+1} (float2 gathers).
// ---------------------------------------------------------------------------
__global__ void k_pull1(const float* __restrict__ xt,
                        const int* __restrict__ row_ptr,
                        const int* __restrict__ csr_src,
                        const float* __restrict__ dis,
                        const float* __restrict__ b1,
                        const float* __restrict__ W2,
                        float* __restrict__ h2t, int n) {
    const int node = (int)((blockIdx.x * (long long)blockDim.x + threadIdx.x) >> 5);
    if (node >= n) return;
    const int lane = threadIdx.x & 31;
    const int j    = lane * 2;

    const float dn  = dis[node];
    const int   beg = row_ptr[node];
    const int   end = row_ptr[node + 1];

    // self-loop term: norm = dn*dn
    const float2 self = *(const float2*)(xt + (long long)node * HID + j);
    float a0 = dn * dn * self.x;
    float a1 = dn * dn * self.y;

    for (int base = beg; base < end; base += 32) {
        const int idx   = base + lane;
        const int valid = (idx < end);
        const int   sl  = valid ? csr_src[idx] : 0;
        const float dl  = valid ? dis[sl]      : 0.0f;
        const int chunk = min(32, end - base);
        for (int t = 0; t < chunk; ++t) {
            const int   s = __shfl(sl, t, 32);
            const float w = __shfl(dl, t, 32) * dn;
            const float2 v = *(const float2*)(xt + (long long)s * HID + j);
            a0 = fmaf(w, v.x, a0);
            a1 = fmaf(w, v.y, a1);
        }
    }

    // layer-2 transform: relu(acc + b1) @ W2   (W2 is [64][2] row-major)
    const float r0 = fmaxf(a0 + b1[j],     0.0f);
    const float r1 = fmaxf(a1 + b1[j + 1], 0.0f);
    float p0 = r0 * W2[j * 2 + 0] + r1 * W2[(j + 1) * 2 + 0];
    float p1 = r0 * W2[j * 2 + 1] + r1 * W2[(j + 1) * 2 + 1];
    #pragma unroll
    for (int off = 16; off > 0; off >>= 1) {
        p0 += __shfl_xor(p0, off, 32);
        p1 += __shfl_xor(p1, off, 32);
    }
    if (lane == 0) {
        h2t[node * 2 + 0] = p0;
        h2t[node * 2 + 1] = p1;
    }
}

// ---------------------------------------------------------------------------
// Pull layer 2 + bias + log_softmax (fully fused, thread per node):
//   o = dis[n]^2 * h2t[n] + sum_s dis[s]*dis[n]*h2t[s] + b2 ;  out = log_softmax(o)
// ---------------------------------------------------------------------------
__global__ void k_pull2(const float* __restrict__ h2t,
                        const int* __restrict__ row_ptr,
                        const int* __restrict__ csr_src,
                        const float* __restrict__ dis,
                        const float* __restrict__ b2,
                        float* __restrict__ out, int n) {
    const int node = blockIdx.x * blockDim.x + threadIdx.x;
    if (node >= n) return;
    const float dn = dis[node];

    float o0 = dn * dn * h2t[node * 2 + 0];
    float o1 = dn * dn * h2t[node * 2 + 1];

    const int beg = row_ptr[node];
    const int end = row_ptr[node + 1];
    for (int i = beg; i < end; ++i) {
        const int   s = csr_src[i];
        const float w = dis[s] * dn;
        o0 = fmaf(w, h2t[s * 2 + 0], o0);
        o1 = fmaf(w, h2t[s * 2 + 1], o1);
    }
    o0 += b2[0];
    o1 += b2[1];

    const float m   = fmaxf(o0, o1);
    const float lse = m + logf(expf(o0 - m) + expf(o1 - m));
    out[node * 2 + 0] = o0 - lse;
    out[node * 2 + 1] = o1 - lse;
}

// ---------------------------------------------------------------------------
extern "C" void kernel_launch(void* const* d_in, const int* in_sizes, int n_in,
                              void* d_out, int out_size, void* d_ws, size_t ws_size,
                              hipStream_t stream) {
    const float* x  = (const float*)d_in[0];
    const int*   ei = (const int*)  d_in[1];
    const float* W1 = (const float*)d_in[2];
    const float* b1 = (const float*)d_in[3];
    const float* W2 = (const float*)d_in[4];
    const float* b2 = (const float*)d_in[5];
    float* out = (float*)d_out;

    const int N = in_sizes[0] / IN_DIM;   // 100000
    const int E = in_sizes[1] / 2;        // 1600000

    // workspace carve-up (256B aligned)
    char* ws = (char*)d_ws;
    size_t off = 0;
    auto carve = [&](size_t bytes) -> char* {
        char* p = ws + off;
        off += (bytes + 255) & ~(size_t)255;
        return p;
    };
    int*   cnt     = (int*)  carve((size_t)N * sizeof(int));          //  0.4 MB
    int*   row_ptr = (int*)  carve((size_t)(N + 1) * sizeof(int));    //  0.4 MB
    int*   pos     = (int*)  carve((size_t)N * sizeof(int));          //  0.4 MB
    int*   csr_src = (int*)  carve((size_t)E * sizeof(int));          //  6.4 MB
    float* dis     = (float*)carve((size_t)N * sizeof(float));        //  0.4 MB
    float* xt      = (float*)carve((size_t)N * HID * sizeof(float));  // 25.6 MB
    float* h2t     = (float*)carve((size_t)N * 2 * sizeof(float));    //  0.8 MB
    (void)ws_size;

    const int B = 256;

    // --- CSR build + normalization ---
    k_zero_cnt<<<(N + B - 1) / B, B, 0, stream>>>(cnt, N);
    k_count   <<<(E + B - 1) / B, B, 0, stream>>>(ei, cnt, E);
    k_scan    <<<1, SCAN_T, 0, stream>>>(cnt, row_ptr, N);
    k_prep    <<<(N + B - 1) / B, B, 0, stream>>>(cnt, row_ptr, dis, pos, N);
    k_scatter <<<(E + B - 1) / B, B, 0, stream>>>(ei, pos, csr_src, E);

    // --- layer 1 dense transform (WMMA) ---
    const int nTiles = N / 16;                       // N = 100000 -> 6250
    const long long gemmThreads = (long long)nTiles * 32;
    k_gemm1_wmma<<<(unsigned)((gemmThreads + B - 1) / B), B, 0, stream>>>(
        x, W1, xt, nTiles);

    // --- layer 1 pull-aggregate fused with layer 2 transform ---
    const long long p1Threads = (long long)N * 32;   // one wave per node
    k_pull1<<<(unsigned)((p1Threads + B - 1) / B), B, 0, stream>>>(
        xt, row_ptr, csr_src, dis, b1, W2, h2t, N);

    // --- layer 2 pull-aggregate + bias + log_softmax ---
    k_pull2<<<(N + B - 1) / B, B, 0, stream>>>(
        h2t, row_ptr, csr_src, dis, b2, out, N);
}